// ModelWrapper_72249939853346
// MI455X (gfx1250) — compile-verified
//
#include <hip/hip_runtime.h>

// ---------------- problem dims ----------------
constexpr int kB     = 64;
constexpr int kSvit  = 257;
constexpr int kSq    = 32;
constexpr int kL     = 512;
constexpr int kIMG   = 1408;
constexpr int kHID   = 768;
constexpr int kC     = 30000;       // 16 * 1875
constexpr int kHclub = 32;
constexpr int kHmine = 64;
constexpr int kCAT   = kIMG + kHID; // 2176

constexpr int kF4     = kIMG / 4;   // 352 float4 per vit row
constexpr int kQ4     = kHID / 4;   // 192 float4 per q row
constexpr int kChunk  = 16;         // tokens per partial-sum chunk
constexpr int kNChunk = (kSvit + kChunk - 1) / kChunk; // 17
constexpr int kCT     = kC / 16;    // 1875 column tiles
constexpr int kTiles  = kCT * (kB / 16); // 7500 club tiles

typedef float v2f __attribute__((ext_vector_type(2)));
typedef float v4f __attribute__((ext_vector_type(4)));
typedef float v8f __attribute__((ext_vector_type(8)));

// V_WMMA_F32_16X16X4_F32: D(16x16,f32) = A(16x4,f32) * B(4x16,f32) + C
// A per lane: a.x=A[l&15][2*(l>>4)], a.y=A[l&15][2*(l>>4)+1]
// B per lane: b.x=B[2*(l>>4)][l&15], b.y=B[2*(l>>4)+1][l&15]
// D per lane: d[r] = D[r + 8*(l>>4)][l&15]
__device__ __forceinline__ v8f wmma_f32_16x16x4(v2f a, v2f b, v8f c) {
  return __builtin_amdgcn_wmma_f32_16x16x4_f32(false, a, false, b, (short)0, c,
                                               false, false);
}

// ---------------- workspace layout (floats) ----------------
constexpr size_t OFF_YH   = 0;                                   // B*C counts
constexpr size_t OFF_IZYP = OFF_YH   + (size_t)kB * kC;          // kTiles (pad)
constexpr size_t OFF_XV   = OFF_IZYP + 7504;                     // B*IMG
constexpr size_t OFF_XQ   = OFF_XV   + (size_t)kB * kIMG;        // B*HID
constexpr size_t OFF_HMU  = OFF_XQ   + (size_t)kB * kHID;        // B*Hclub
constexpr size_t OFF_HLV  = OFF_HMU  + (size_t)kB * kHclub;      // B*Hclub
constexpr size_t OFF_H0   = OFF_HLV  + (size_t)kB * kHclub;      // B*CAT
constexpr size_t OFF_H1   = OFF_H0   + (size_t)kB * kCAT;        // B*CAT
constexpr size_t OFF_U0   = OFF_H1   + (size_t)kB * kCAT;        // B*Hmine
constexpr size_t OFF_U1   = OFF_U0   + (size_t)kB * kHmine;      // B*Hmine
constexpr size_t OFF_VP   = OFF_U1   + (size_t)kB * kHmine;      // partials
// OFF_VP size = kNChunk * kB * kF4 * 4 floats

// ---------------- kernels ----------------
__global__ void zero_kernel(float* __restrict__ p, int n) {
  int i = blockIdx.x * blockDim.x + threadIdx.x;
  if (i < n) p[i] = 0.0f;
}

// Phase 1: per-chunk partial column sums of vit, float4 + non-temporal loads.
// grid (B*kF4/256, kNChunk). Thread (i,z): sum rows [z*16, min(257,z*16+16)).
__global__ void mean_vit_part_kernel(const float* __restrict__ vit,
                                     float* __restrict__ vpart) {
  int i = blockIdx.x * blockDim.x + threadIdx.x;   // 0 .. B*kF4
  if (i >= kB * kF4) return;
  int z  = blockIdx.y;
  int b  = i / kF4, f4 = i % kF4;
  int s0 = z * kChunk;
  int s1 = (s0 + kChunk < kSvit) ? (s0 + kChunk) : kSvit;
  const v4f* p = (const v4f*)(vit + (size_t)b * kSvit * kIMG) +
                 (size_t)s0 * kF4 + f4;
  v4f acc = {0.0f, 0.0f, 0.0f, 0.0f};
  for (int s = s0; s < s1; ++s) {
    acc += __builtin_nontemporal_load(p);
    p += kF4;
  }
  ((v4f*)vpart)[(size_t)z * kB * kF4 + i] = acc;   // stays hot in L2
}

// Phase 2: fixed-order reduce of the 17 chunk partials -> xv (deterministic).
__global__ void mean_vit_reduce_kernel(const float* __restrict__ vpart,
                                       float* __restrict__ xv) {
  int i = blockIdx.x * blockDim.x + threadIdx.x;
  if (i >= kB * kF4) return;
  const v4f* p = (const v4f*)vpart + i;
  v4f acc = {0.0f, 0.0f, 0.0f, 0.0f};
  for (int z = 0; z < kNChunk; ++z) acc += p[(size_t)z * kB * kF4];
  ((v4f*)xv)[i] = acc * (1.0f / (float)kSvit);
}

// xq[b][:] = mean_s qformer[b][s][:], float4 + non-temporal loads.
__global__ void mean_q_kernel(const float* __restrict__ q,
                              float* __restrict__ xq) {
  int i = blockIdx.x * blockDim.x + threadIdx.x;   // 0 .. B*kQ4
  if (i >= kB * kQ4) return;
  int b = i / kQ4, f4 = i % kQ4;
  const v4f* p = (const v4f*)(q + (size_t)b * kSq * kHID) + f4;
  v4f acc = {0.0f, 0.0f, 0.0f, 0.0f};
  for (int s = 0; s < kSq; ++s) {
    acc += __builtin_nontemporal_load(p);
    p += kQ4;
  }
  ((v4f*)xq)[i] = acc * (1.0f / (float)kSq);
}

// y_hist counts via float atomics (integer-valued adds -> order-invariant).
__global__ void hist_kernel(const int* __restrict__ label,
                            float* __restrict__ yh) {
  int i = blockIdx.x * blockDim.x + threadIdx.x;
  if (i >= kB * kL) return;
  int lab = label[i];
  if (lab == -100) lab = 0;
  int b = i / kL;
  atomicAdd(&yh[(size_t)b * kC + lab], 1.0f);
}

// h = relu(xq @ W1 + b1), shape (64,32). blockIdx.x: 0 -> mu path, 1 -> lv.
// 8 waves, one 16x16 output tile per wave (4 row-tiles x 2 col-tiles), K=768.
__global__ void club_hidden_kernel(const float* __restrict__ xq,
                                   const float* __restrict__ W_mu1,
                                   const float* __restrict__ b_mu1,
                                   const float* __restrict__ W_lv1,
                                   const float* __restrict__ b_lv1,
                                   float* __restrict__ h_mu,
                                   float* __restrict__ h_lv) {
  const float* W    = (blockIdx.x == 0) ? W_mu1 : W_lv1;
  const float* bias = (blockIdx.x == 0) ? b_mu1 : b_lv1;
  float* out        = (blockIdx.x == 0) ? h_mu  : h_lv;

  int wave = threadIdx.x >> 5;    // 0..7
  int lane = threadIdx.x & 31;
  int tm = wave >> 1;             // row tile 0..3
  int tn = wave & 1;              // col tile 0..1
  int kh = lane >> 4;             // K half-select
  int rowA = tm * 16 + (lane & 15);
  int n    = tn * 16 + (lane & 15);

  v8f acc = {};
  for (int k0 = 0; k0 < kHID; k0 += 4) {
    int ka = k0 + kh * 2;
    v2f a, b;
    a.x = xq[(size_t)rowA * kHID + ka];
    a.y = xq[(size_t)rowA * kHID + ka + 1];
    b.x = W[(size_t)ka * kHclub + n];
    b.y = W[(size_t)(ka + 1) * kHclub + n];
    acc = wmma_f32_16x16x4(a, b, acc);
  }
  float bn = bias[n];
#pragma unroll
  for (int r = 0; r < 8; ++r) {
    int row = tm * 16 + r + kh * 8;
    out[(size_t)row * kHclub + n] = fmaxf(acc[r] + bn, 0.0f);
  }
}

// Fused CLUB: mu/logvar 16x16 tile GEMMs (K=32, WMMA) + tanh/exp + (pos-neg).
// 8 waves per block, one column tile per wave; per-tile partial -> izy_part
// (reduced in fixed order later => deterministic). Never materializes (B,C).
__global__ void club_main_kernel(const float* __restrict__ h_mu,
                                 const float* __restrict__ h_lv,
                                 const float* __restrict__ W_mu2,
                                 const float* __restrict__ b_mu2,
                                 const float* __restrict__ W_lv2,
                                 const float* __restrict__ b_lv2,
                                 const float* __restrict__ yh,
                                 const int* __restrict__ perm,
                                 float* __restrict__ izy_part) {
  int wave = threadIdx.x >> 5;
  int lane = threadIdx.x & 31;
  int ct = blockIdx.x * 8 + wave;   // column tile (wave-uniform)
  int bt = blockIdx.y;              // row tile 0..3
  if (ct >= kCT) return;            // uniform per wave: EXEC stays all-ones

  int kh = lane >> 4;
  int rowA = bt * 16 + (lane & 15);
  int col  = ct * 16 + (lane & 15);

  v8f cmu = {}, clv = {};
  for (int k0 = 0; k0 < kHclub; k0 += 4) {
    int ka = k0 + kh * 2;
    v2f a, b;
    a.x = h_mu[(size_t)rowA * kHclub + ka];
    a.y = h_mu[(size_t)rowA * kHclub + ka + 1];
    b.x = W_mu2[(size_t)ka * kC + col];
    b.y = W_mu2[(size_t)(ka + 1) * kC + col];
    cmu = wmma_f32_16x16x4(a, b, cmu);
    a.x = h_lv[(size_t)rowA * kHclub + ka];
    a.y = h_lv[(size_t)rowA * kHclub + ka + 1];
    b.x = W_lv2[(size_t)ka * kC + col];
    b.y = W_lv2[(size_t)(ka + 1) * kC + col];
    clv = wmma_f32_16x16x4(a, b, clv);
  }

  float bm = b_mu2[col], bl = b_lv2[col];
  const float invL = 1.0f / (float)kL;
  float part = 0.0f;
#pragma unroll
  for (int r = 0; r < 8; ++r) {
    int b  = bt * 16 + r + kh * 8;     // global batch row of this element
    float mu = cmu[r] + bm;
    float lv = tanhf(clv[r] + bl);
    float iv = 1.0f / (expf(lv) + 1e-6f);
    float yb = yh[(size_t)b * kC + col] * invL;
    float yp = yh[(size_t)perm[b] * kC + col] * invL;
    float db = mu - yb, dp = mu - yp;
    part += (dp * dp - db * db) * iv;  // (pos - neg) contribution
  }
  for (int off = 16; off > 0; off >>= 1) part += __shfl_xor(part, off, 32);
  if (lane == 0) izy_part[bt * kCT + ct] = part;
}

// H0[b] = [xv[b], xq[b]] ; H1[b] = [xv[b], xq[rand[b]]]
__global__ void concat_kernel(const float* __restrict__ xv,
                              const float* __restrict__ xq,
                              const int* __restrict__ rnd,
                              float* __restrict__ H0,
                              float* __restrict__ H1) {
  int i = blockIdx.x * blockDim.x + threadIdx.x;
  if (i >= kB * kCAT) return;
  int b = i / kCAT, k = i % kCAT;
  float v0, v1;
  if (k < kIMG) {
    v0 = v1 = xv[(size_t)b * kIMG + k];
  } else {
    v0 = xq[(size_t)b * kHID + (k - kIMG)];
    v1 = xq[(size_t)rnd[b] * kHID + (k - kIMG)];
  }
  H0[i] = v0;
  H1[i] = v1;
}

// U = relu(H @ W_t1 + b_t1), (64,2176)x(2176,64). 32 one-wave blocks:
// blocks 0..15 -> H0/U0 tiles, 16..31 -> H1/U1 tiles. K=2176 -> 544 WMMAs.
__global__ void mine_gemm_kernel(const float* __restrict__ H0,
                                 const float* __restrict__ H1,
                                 const float* __restrict__ W_t1,
                                 const float* __restrict__ b_t1,
                                 float* __restrict__ U0,
                                 float* __restrict__ U1) {
  int z = blockIdx.x;
  const float* H = (z < 16) ? H0 : H1;
  float* U       = (z < 16) ? U0 : U1;
  int t  = z & 15;
  int tm = t >> 2, tn = t & 3;
  int lane = threadIdx.x;
  int kh = lane >> 4;
  int rowA = tm * 16 + (lane & 15);
  int n    = tn * 16 + (lane & 15);

  v8f acc = {};
  for (int k0 = 0; k0 < kCAT; k0 += 4) {
    int ka = k0 + kh * 2;
    v2f a, b;
    a.x = H[(size_t)rowA * kCAT + ka];
    a.y = H[(size_t)rowA * kCAT + ka + 1];
    b.x = W_t1[(size_t)ka * kHmine + n];
    b.y = W_t1[(size_t)(ka + 1) * kHmine + n];
    acc = wmma_f32_16x16x4(a, b, acc);
  }
  float bn = b_t1[n];
#pragma unroll
  for (int r = 0; r < 8; ++r) {
    int row = tm * 16 + r + kh * 8;
    U[(size_t)row * kHmine + n] = fmaxf(acc[r] + bn, 0.0f);
  }
}

// Final: t0/t1 = U @ W_t2 + b_t2; I_xz = mean(t0) - (lse(t1) - log B);
// I_zy = (fixed-order sum of izy_part)/(2B); out = I_zy - 0.1*I_xz.
__global__ void final_kernel(const float* __restrict__ U0,
                             const float* __restrict__ U1,
                             const float* __restrict__ W_t2,
                             const float* __restrict__ b_t2,
                             const float* __restrict__ izy_part,
                             float* __restrict__ out) {
  __shared__ float t0s[kB], t1s[kB], reds[kB];
  int tid = threadIdx.x;            // blockDim.x == kB == 64
  float a0 = 0.0f, a1 = 0.0f;
  for (int j = 0; j < kHmine; ++j) {
    float w = W_t2[j];
    a0 += U0[(size_t)tid * kHmine + j] * w;
    a1 += U1[(size_t)tid * kHmine + j] * w;
  }
  t0s[tid] = a0 + b_t2[0];
  t1s[tid] = a1 + b_t2[0];
  float ps = 0.0f;
  for (int i = tid; i < kTiles; i += kB) ps += izy_part[i];
  reds[tid] = ps;
  __syncthreads();
  if (tid == 0) {
    float izy_sum = 0.0f;
    for (int i = 0; i < kB; ++i) izy_sum += reds[i];
    float s0 = 0.0f, mx = -3.402823466e38f;
    for (int i = 0; i < kB; ++i) {
      s0 += t0s[i];
      mx = fmaxf(mx, t1s[i]);
    }
    float se = 0.0f;
    for (int i = 0; i < kB; ++i) se += expf(t1s[i] - mx);
    float lse = mx + logf(se);
    float I_xz = s0 / (float)kB - (lse - logf((float)kB));
    float I_zy = izy_sum / (2.0f * (float)kB);
    out[0] = I_zy - 0.1f * I_xz;
  }
}

// ---------------- host launcher ----------------
extern "C" void kernel_launch(void* const* d_in, const int* in_sizes, int n_in,
                              void* d_out, int out_size, void* d_ws,
                              size_t ws_size, hipStream_t stream) {
  (void)in_sizes; (void)n_in; (void)out_size; (void)ws_size;
  const float* vit    = (const float*)d_in[0];
  const float* qfrm   = (const float*)d_in[1];
  const int*   label  = (const int*)d_in[2];
  const int*   perm   = (const int*)d_in[3];
  const int*   rnd    = (const int*)d_in[4];
  const float* W_mu1  = (const float*)d_in[5];
  const float* b_mu1  = (const float*)d_in[6];
  const float* W_mu2  = (const float*)d_in[7];
  const float* b_mu2  = (const float*)d_in[8];
  const float* W_lv1  = (const float*)d_in[9];
  const float* b_lv1  = (const float*)d_in[10];
  const float* W_lv2  = (const float*)d_in[11];
  const float* b_lv2  = (const float*)d_in[12];
  const float* W_t1   = (const float*)d_in[13];
  const float* b_t1   = (const float*)d_in[14];
  const float* W_t2   = (const float*)d_in[15];
  const float* b_t2   = (const float*)d_in[16];

  float* ws   = (float*)d_ws;
  float* yh   = ws + OFF_YH;
  float* izyp = ws + OFF_IZYP;
  float* xv   = ws + OFF_XV;
  float* xq   = ws + OFF_XQ;
  float* hmu  = ws + OFF_HMU;
  float* hlv  = ws + OFF_HLV;
  float* H0   = ws + OFF_H0;
  float* H1   = ws + OFF_H1;
  float* U0   = ws + OFF_U0;
  float* U1   = ws + OFF_U1;
  float* vp   = ws + OFF_VP;
  float* out  = (float*)d_out;

  // 1) zero hist counts
  int zn = kB * kC;
  zero_kernel<<<(zn + 255) / 256, 256, 0, stream>>>(yh, zn);
  // 2) vit mean: chunked partials (high MLP for 93 MB stream), then reduce
  mean_vit_part_kernel<<<dim3((kB * kF4 + 255) / 256, kNChunk), 256, 0,
                         stream>>>(vit, vp);
  mean_vit_reduce_kernel<<<(kB * kF4 + 255) / 256, 256, 0, stream>>>(vp, xv);
  // 3) qformer mean
  mean_q_kernel<<<(kB * kQ4 + 255) / 256, 256, 0, stream>>>(qfrm, xq);
  // 4) label histogram
  hist_kernel<<<(kB * kL + 255) / 256, 256, 0, stream>>>(label, yh);
  // 5) CLUB hidden layers (WMMA f32)
  club_hidden_kernel<<<2, 256, 0, stream>>>(xq, W_mu1, b_mu1, W_lv1, b_lv1,
                                            hmu, hlv);
  // 6) fused CLUB wide GEMM + pos/neg reduction (WMMA f32)
  club_main_kernel<<<dim3((kCT + 7) / 8, kB / 16), 256, 0, stream>>>(
      hmu, hlv, W_mu2, b_mu2, W_lv2, b_lv2, yh, perm, izyp);
  // 7) MINE concat inputs
  concat_kernel<<<(kB * kCAT + 255) / 256, 256, 0, stream>>>(xv, xq, rnd, H0,
                                                             H1);
  // 8) MINE layer-1 GEMMs (WMMA f32)
  mine_gemm_kernel<<<32, 32, 0, stream>>>(H0, H1, W_t1, b_t1, U0, U1);
  // 9) final combine -> scalar (deterministic fixed-order reductions)
  final_kernel<<<1, kB, 0, stream>>>(U0, U1, W_t2, b_t2, izyp, out);
}